// SolveBoxQP_66992899883233
// MI455X (gfx1250) — compile-verified
//
#include <hip/hip_runtime.h>
#include <math.h>

// CDNA5 / gfx1250: wave32, WMMA f32 16x16x4 fragments
typedef __attribute__((ext_vector_type(2))) float v2f;
typedef __attribute__((ext_vector_type(8))) float v8f;

namespace {
constexpr int   kB       = 4096;
constexpr int   kNx      = 64;
constexpr int   kNeq     = 16;
constexpr int   kNt      = 80;     // kNx + kNeq
constexpr float kRho     = 1.0f;
constexpr int   kIters   = 50;
constexpr float kEpsAbs  = 1e-3f;
constexpr float kEpsRel  = 1e-3f;

// workspace layout (units: floats)
constexpr size_t kWsMinv = 0;                               // kB * 6400 (fragment-ordered M^-1)
constexpr size_t kWsX    = kWsMinv + (size_t)kB * 6400;     // kB * 64
constexpr size_t kWsZ    = kWsX    + (size_t)kB * kNx;      // kB * 64
constexpr size_t kWsU    = kWsZ    + (size_t)kB * kNx;      // kB * 64
constexpr size_t kWsNus  = kWsU    + (size_t)kB * kNx;      // kB * 16
constexpr size_t kWsSums = kWsNus  + (size_t)kB * kNeq;     // 5 accumulators
constexpr size_t kWsDone = kWsSums + 8;                     // 1 int flag
constexpr size_t kWsEnd  = kWsDone + 8;
}

// ---------------------------------------------------------------------------
// Zero ADMM state + accumulators + done flag (ws is poisoned by the harness).
// ---------------------------------------------------------------------------
__global__ void admm_init(float* ws) {
  size_t n = kWsEnd - kWsX;
  size_t i = (size_t)blockIdx.x * blockDim.x + threadIdx.x;
  if (i < n) ws[kWsX + i] = 0.0f;
}

// ---------------------------------------------------------------------------
// Per batch: build KKT matrix M = [[Q + rho*I, A^T], [A, 0]] (80x80) in LDS,
// Gauss-Jordan invert with partial pivoting, store M^-1 in WMMA-fragment
// order so the iteration kernel's A-operand loads are fully coalesced.
// Fragment order: for rowblock rb (5) x chunk ch (20): 32 lanes x v2f.
//   lane = (r%16) + 16*(c%4 >= 2), elem = (c%4)&1
// ---------------------------------------------------------------------------
__global__ __launch_bounds__(256) void build_invert(const float* __restrict__ Q,
                                                    const float* __restrict__ A,
                                                    float* __restrict__ ws) {
  const int bt  = blockIdx.x;
  const int tid = threadIdx.x;
  __shared__ float M[kNt][2 * kNt + 1];   // 80 x 161 (padded), ~52 KB
  __shared__ float fcol[kNt];
  __shared__ int   s_piv;
  __shared__ float s_pinv;

  const float* Qb = Q + (size_t)bt * kNx * kNx;
  const float* Ab = A + (size_t)bt * kNeq * kNx;

  // Build augmented [M | I]
  for (int idx = tid; idx < kNt * 2 * kNt; idx += 256) {
    int i = idx / (2 * kNt), j = idx % (2 * kNt);
    float v;
    if (j < kNt) {
      if (i < kNx && j < kNx)       v = Qb[i * kNx + j] + (i == j ? kRho : 0.0f);
      else if (i < kNx)             v = Ab[(j - kNx) * kNx + i];     // A^T
      else if (j < kNx)             v = Ab[(i - kNx) * kNx + j];     // A
      else                          v = 0.0f;
    } else {
      v = ((j - kNt) == i) ? 1.0f : 0.0f;
    }
    M[i][j] = v;
  }
  __syncthreads();

  for (int k = 0; k < kNt; ++k) {
    if (tid == 0) {
      int pr = k; float pv = fabsf(M[k][k]);
      for (int i = k + 1; i < kNt; ++i) {
        float v = fabsf(M[i][k]);
        if (v > pv) { pv = v; pr = i; }
      }
      s_piv = pr;
    }
    __syncthreads();
    const int pr = s_piv;
    if (pr != k) {
      for (int j = tid; j < 2 * kNt; j += 256) {
        float t = M[k][j]; M[k][j] = M[pr][j]; M[pr][j] = t;
      }
    }
    __syncthreads();
    if (tid == 0) s_pinv = 1.0f / M[k][k];
    __syncthreads();
    const float pinv = s_pinv;
    for (int j = tid; j < 2 * kNt; j += 256) M[k][j] *= pinv;
    __syncthreads();
    for (int i = tid; i < kNt; i += 256) fcol[i] = M[i][k];
    __syncthreads();
    for (int idx = tid; idx < kNt * 2 * kNt; idx += 256) {
      int i = idx / (2 * kNt), j = idx % (2 * kNt);
      if (i != k) M[i][j] -= fcol[i] * M[k][j];
    }
    __syncthreads();
  }

  // Emit right half (= M^-1) in WMMA A-fragment order.
  float* mb = ws + kWsMinv + (size_t)bt * 6400;
  for (int idx = tid; idx < kNt * kNt; idx += 256) {
    int r = idx / kNt, c = idx % kNt;
    int rb = r >> 4, rr = r & 15, ch = c >> 2, cc = c & 3;
    int lane = rr + ((cc >> 1) << 4);
    int elem = cc & 1;
    mb[((size_t)(rb * 20 + ch) * 32 + lane) * 2 + elem] = M[r][kNt + c];
  }
}

// ---------------------------------------------------------------------------
// One ADMM iteration. One wave32 per batch. Matvec xv = M^-1 * rhs via
// v_wmma_f32_16x16x4_f32 (5 row-blocks x 20 K-chunks, rhs broadcast over N).
// ---------------------------------------------------------------------------
__global__ __launch_bounds__(32) void admm_step(const float* __restrict__ p,
                                                const float* __restrict__ beq,
                                                const float* __restrict__ lb,
                                                const float* __restrict__ ub,
                                                float* __restrict__ ws) {
  if (*(volatile const int*)(ws + kWsDone)) return;   // converged: no-op launch

  const int bt   = blockIdx.x;
  const int lane = threadIdx.x;

  __shared__ float rhs[kNt];
  __shared__ float xv[kNt];

  const float* pb  = p   + (size_t)bt * kNx;
  const float* bb  = beq + (size_t)bt * kNeq;
  const float* lbb = lb  + (size_t)bt * kNx;
  const float* ubb = ub  + (size_t)bt * kNx;
  float* xs = ws + kWsX + (size_t)bt * kNx;
  float* zs = ws + kWsZ + (size_t)bt * kNx;
  float* us = ws + kWsU + (size_t)bt * kNx;

  // y = z - u ; rhs = [-p + rho*y ; b]
  float z0 = zs[lane],      z1 = zs[lane + 32];
  float u0 = us[lane],      u1 = us[lane + 32];
  float y0 = z0 - u0,       y1 = z1 - u1;
  rhs[lane]      = -pb[lane]      + kRho * y0;
  rhs[lane + 32] = -pb[lane + 32] + kRho * y1;
  if (lane < kNeq) rhs[kNx + lane] = bb[lane];
  __syncthreads();

  const float* mb = ws + kWsMinv + (size_t)bt * 6400;
  __builtin_prefetch(mb + (size_t)lane * 200, 0, 1);  // warm this batch's M^-1 (L2-resident)

  v8f acc0 = {0,0,0,0,0,0,0,0};
  v8f acc1 = acc0, acc2 = acc0, acc3 = acc0, acc4 = acc0;

  for (int ch = 0; ch < 20; ++ch) {
    // B fragment (4x16): K split lanes 0-15 -> K0,K1 ; lanes 16-31 -> K2,K3;
    // broadcast rhs chunk across all 16 N columns.
    v2f bf;
    bf.x = rhs[4 * ch + (lane < 16 ? 0 : 2)];
    bf.y = rhs[4 * ch + (lane < 16 ? 1 : 3)];
    const v2f* ap = (const v2f*)(mb + (size_t)ch * 64);  // chunk-major within rowblock stride
    // A fragments: fragment-ordered storage => one coalesced b64 load per rowblock
    v2f a0 = *(const v2f*)(mb + ((size_t)(0 * 20 + ch) * 32 + lane) * 2);
    v2f a1 = *(const v2f*)(mb + ((size_t)(1 * 20 + ch) * 32 + lane) * 2);
    v2f a2 = *(const v2f*)(mb + ((size_t)(2 * 20 + ch) * 32 + lane) * 2);
    v2f a3 = *(const v2f*)(mb + ((size_t)(3 * 20 + ch) * 32 + lane) * 2);
    v2f a4 = *(const v2f*)(mb + ((size_t)(4 * 20 + ch) * 32 + lane) * 2);
    (void)ap;
    acc0 = __builtin_amdgcn_wmma_f32_16x16x4_f32(false, a0, false, bf, (short)0, acc0, false, false);
    acc1 = __builtin_amdgcn_wmma_f32_16x16x4_f32(false, a1, false, bf, (short)0, acc1, false, false);
    acc2 = __builtin_amdgcn_wmma_f32_16x16x4_f32(false, a2, false, bf, (short)0, acc2, false, false);
    acc3 = __builtin_amdgcn_wmma_f32_16x16x4_f32(false, a3, false, bf, (short)0, acc3, false, false);
    acc4 = __builtin_amdgcn_wmma_f32_16x16x4_f32(false, a4, false, bf, (short)0, acc4, false, false);
  }

  // Extract xv: C/D layout -> lane N=0 (lanes 0 and 16) holds rows v (+8).
  if ((lane & 15) == 0) {
    int half = (lane >= 16) ? 8 : 0;
#pragma unroll
    for (int v = 0; v < 8; ++v) {
      xv[ 0 + half + v] = acc0[v];
      xv[16 + half + v] = acc1[v];
      xv[32 + half + v] = acc2[v];
      xv[48 + half + v] = acc3[v];
      xv[64 + half + v] = acc4[v];
    }
  }
  __syncthreads();

  if (lane < kNeq) (ws + kWsNus)[(size_t)bt * kNeq + lane] = xv[kNx + lane];

  // x/z/u update + per-batch norm partials (2 elements per lane)
  float pr = 0.f, ds = 0.f, xn = 0.f, zn = 0.f, yn = 0.f;
  {
    float x  = xv[lane];
    float znw = fminf(fmaxf(x + u0, lbb[lane]), ubb[lane]);
    float r  = x - znw;
    float s  = kRho * (znw - z0);
    float un = u0 + r;
    xs[lane] = x; zs[lane] = znw; us[lane] = un;
    pr += r * r; ds += s * s; xn += x * x; zn += znw * znw; yn += y0 * y0;
  }
  {
    float x  = xv[lane + 32];
    float znw = fminf(fmaxf(x + u1, lbb[lane + 32]), ubb[lane + 32]);
    float r  = x - znw;
    float s  = kRho * (znw - z1);
    float un = u1 + r;
    xs[lane + 32] = x; zs[lane + 32] = znw; us[lane + 32] = un;
    pr += r * r; ds += s * s; xn += x * x; zn += znw * znw; yn += y1 * y1;
  }
#pragma unroll
  for (int off = 16; off > 0; off >>= 1) {
    pr += __shfl_down(pr, off, 32);
    ds += __shfl_down(ds, off, 32);
    xn += __shfl_down(xn, off, 32);
    zn += __shfl_down(zn, off, 32);
    yn += __shfl_down(yn, off, 32);
  }
  if (lane == 0) {
    atomicAdd(ws + kWsSums + 0, sqrtf(pr));
    atomicAdd(ws + kWsSums + 1, sqrtf(ds));
    atomicAdd(ws + kWsSums + 2, sqrtf(xn));
    atomicAdd(ws + kWsSums + 3, sqrtf(zn));
    atomicAdd(ws + kWsSums + 4, sqrtf(yn));
  }
}

// ---------------------------------------------------------------------------
// Global (batch-mean) convergence test, mirroring the reference while_loop.
// ---------------------------------------------------------------------------
__global__ void admm_check(float* ws) {
  int* done = (int*)(ws + kWsDone);
  if (*done) return;
  const float invB = 1.0f / (float)kB;
  float prim = ws[kWsSums + 0] * invB;
  float dual = ws[kWsSums + 1] * invB;
  float xn   = ws[kWsSums + 2] * invB;
  float zn   = ws[kWsSums + 3] * invB;
  float yn   = ws[kWsSums + 4] * invB;
  const float sqrt_nx = 8.0f;  // sqrt(64)
  float tolp = kEpsAbs * sqrt_nx + kEpsRel * fmaxf(xn, zn);
  float told = kEpsAbs * sqrt_nx + kEpsRel * yn;
  if (prim < tolp && dual < told) *done = 1;
#pragma unroll
  for (int i = 0; i < 5; ++i) ws[kWsSums + i] = 0.0f;
}

// ---------------------------------------------------------------------------
// Emit (x, z, u, lams, nus) concatenated flat.
// ---------------------------------------------------------------------------
__global__ void write_out(const float* __restrict__ ws, float* __restrict__ out) {
  size_t i = (size_t)blockIdx.x * blockDim.x + threadIdx.x;
  if (i >= (size_t)kB * kNx) return;
  int bt = (int)(i >> 6), j = (int)(i & 63);
  float x = ws[kWsX + i], z = ws[kWsZ + i], u = ws[kWsU + i];
  const size_t oX = 0, oZ = (size_t)kB * kNx, oU = 2 * oZ, oL = 3 * oZ,
               oN = oL + (size_t)kB * 2 * kNx;
  out[oX + i] = x;
  out[oZ + i] = z;
  out[oU + i] = u;
  float lam = kRho * u;
  out[oL + (size_t)bt * 128 + j]      = fmaxf(-lam, 0.0f);
  out[oL + (size_t)bt * 128 + 64 + j] = fmaxf( lam, 0.0f);
  if (j < kNeq) out[oN + (size_t)bt * kNeq + j] = ws[kWsNus + (size_t)bt * kNeq + j];
}

extern "C" void kernel_launch(void* const* d_in, const int* in_sizes, int n_in,
                              void* d_out, int out_size, void* d_ws, size_t ws_size,
                              hipStream_t stream) {
  (void)in_sizes; (void)n_in; (void)out_size; (void)ws_size;
  const float* Q  = (const float*)d_in[0];
  const float* p  = (const float*)d_in[1];
  const float* A  = (const float*)d_in[2];
  const float* b  = (const float*)d_in[3];
  const float* lb = (const float*)d_in[4];
  const float* ub = (const float*)d_in[5];
  float* ws  = (float*)d_ws;
  float* out = (float*)d_out;

  const size_t initN = kWsEnd - kWsX;
  admm_init<<<(unsigned)((initN + 255) / 256), 256, 0, stream>>>(ws);
  build_invert<<<kB, 256, 0, stream>>>(Q, A, ws);
  for (int it = 0; it < kIters; ++it) {
    admm_step<<<kB, 32, 0, stream>>>(p, b, lb, ub, ws);
    admm_check<<<1, 1, 0, stream>>>(ws);
  }
  write_out<<<(kB * kNx + 255) / 256, 256, 0, stream>>>(ws, out);
}